// NewAttention_2327872275034
// MI455X (gfx1250) — compile-verified
//
#include <hip/hip_runtime.h>
#include <hip/hip_bf16.h>

// ---------------------------------------------------------------------------
// Gaussian positional attention on MI455X (gfx1250, wave32, WMMA).
//   Y   = values @ W_in^T            (f32 WMMA GEMM, mask applied at store)
//   Att = band(attn) @ Y per (b,h)   (attn computed analytically in regs)
//   out = Att @ W_out^T              (f32 WMMA GEMM)
// attn weights underflow f32 beyond |v-ofs-q| > 14.3, so a 64-wide band
// (vbase = q0-24) is exactly equivalent to the dense einsum in f32.
//
// GEMM wave tile: 32(M) x 64(N), 8 f32 accumulators. Per K=4 step:
//   2 A-fragment loads + 4 W-fragment loads (b64 each) : 8 WMMAs
//   -> 16384 FLOP per 1536 B of wave traffic (~10.7 FLOP/B), all of it
//   L2-resident (72 MB working set << 192 MB L2).
// ---------------------------------------------------------------------------

typedef __attribute__((ext_vector_type(2))) float v2f;
typedef __attribute__((ext_vector_type(8))) float v8f;

static constexpr int Bb = 8, Ll = 1024, Ee = 1024, Hh = 8, Dd = 128;

__device__ __forceinline__ v8f wmma_f32_16x16x4(v2f a, v2f b, v8f c) {
  // D(16x16,f32) = A(16x4,f32) x B(4x16,f32) + C
  return __builtin_amdgcn_wmma_f32_16x16x4_f32(
      /*neg_a=*/false, a, /*neg_b=*/false, b,
      /*c_mod=*/(short)0, c, /*reuse_a=*/false, /*reuse_b=*/false);
}

// C[M,N] = A[M,K] @ W[N,K]^T ; one wave computes a 32(M) x 64(N) tile.
// If MASK: zero output rows where rowmask[row] != 0 (applied at store so
// EXEC stays all-1s around the WMMA ops).
template <bool MASK>
__global__ __launch_bounds__(256) void gemm_xwt(
    const float* __restrict__ A, const float* __restrict__ W,
    float* __restrict__ C, int M, int N, int K,
    const unsigned char* __restrict__ rowmask) {
  const int lane = threadIdx.x & 31;
  const int lp = lane & 15;        // M (for A) / N (for B) position
  const int hf = lane >> 4;        // K half: lanes 16-31 hold K+2,K+3
  const int gw = blockIdx.x * (blockDim.x >> 5) + (threadIdx.x >> 5);
  const int tiles_m = M >> 5;
  const int m0 = (gw % tiles_m) << 5;
  const int n0 = (gw / tiles_m) << 6;

  const float* a0p = A + (size_t)(m0 + lp) * K + 2 * hf;
  const float* a1p = a0p + (size_t)16 * K;
  const float* w0p = W + (size_t)(n0 + lp) * K + 2 * hf;
  const float* w1p = w0p + (size_t)16 * K;
  const float* w2p = w0p + (size_t)32 * K;
  const float* w3p = w0p + (size_t)48 * K;

  v8f c00 = {}, c01 = {}, c02 = {}, c03 = {};
  v8f c10 = {}, c11 = {}, c12 = {}, c13 = {};
  for (int k = 0; k < K; k += 4) {
    v2f a0 = *(const v2f*)(a0p + k);
    v2f a1 = *(const v2f*)(a1p + k);
    v2f b0 = *(const v2f*)(w0p + k);
    v2f b1 = *(const v2f*)(w1p + k);
    v2f b2 = *(const v2f*)(w2p + k);
    v2f b3 = *(const v2f*)(w3p + k);
    c00 = wmma_f32_16x16x4(a0, b0, c00);
    c01 = wmma_f32_16x16x4(a0, b1, c01);
    c02 = wmma_f32_16x16x4(a0, b2, c02);
    c03 = wmma_f32_16x16x4(a0, b3, c03);
    c10 = wmma_f32_16x16x4(a1, b0, c10);
    c11 = wmma_f32_16x16x4(a1, b1, c11);
    c12 = wmma_f32_16x16x4(a1, b2, c12);
    c13 = wmma_f32_16x16x4(a1, b3, c13);
  }

#pragma unroll
  for (int g = 0; g < 8; ++g) {
    // C/D layout: VGPR g holds rows g (lanes 0-15) and g+8 (lanes 16-31)
    const int r0 = m0 + g + 8 * hf;        // M sub-tile 0
    const int r1 = r0 + 16;                // M sub-tile 1
    float keep0 = 1.0f, keep1 = 1.0f;
    if (MASK) {
      keep0 = rowmask[r0] ? 0.0f : 1.0f;
      keep1 = rowmask[r1] ? 0.0f : 1.0f;
    }
    float* crow0 = C + (size_t)r0 * N + n0 + lp;
    float* crow1 = C + (size_t)r1 * N + n0 + lp;
    crow0[0]  = c00[g] * keep0;
    crow0[16] = c01[g] * keep0;
    crow0[32] = c02[g] * keep0;
    crow0[48] = c03[g] * keep0;
    crow1[0]  = c10[g] * keep1;
    crow1[16] = c11[g] * keep1;
    crow1[32] = c12[g] * keep1;
    crow1[48] = c13[g] * keep1;
  }
}

// Att[b, q, h*128 + n] = sum_v attn[h,q,v] * Y[b, v, h*128 + n]
// One wave: (b,h) x 16-row q-tile x 64-col n-group; K = 64-wide exact band.
__global__ __launch_bounds__(256) void gauss_attn_apply(
    const float* __restrict__ V,   // [B, L, E] projected + masked values
    float* __restrict__ O) {       // [B, L, E] attended (head-major cols)
  const int lane = threadIdx.x & 31;
  const int lp = lane & 15;
  const int hf = lane >> 4;
  const int gw = blockIdx.x * (blockDim.x >> 5) + (threadIdx.x >> 5);
  const int ng = gw & 1;           // which 64-col half of the head dim
  const int qt = (gw >> 1) & 63;   // q tile (L/16 = 64)
  const int bh = gw >> 7;          // 0..63
  const int b  = bh >> 3, h = bh & 7;

  // ATTN_OFFSET = {-3,-2,-1,0,0,1,2,3}, ATTN_STD = 1
  const float ofs = (float)((h < 4) ? (h - 3) : (h - 4));
  const int q0 = qt << 4;
  const int vbase = q0 - 24;                 // band covers |v - q - ofs| <= 18
  const int col = h * Dd + ng * 64 + lp;
  const float qf = (float)(q0 + lp);
  const float* Vb = V + (size_t)b * Ll * Ee;

  v8f c0 = {}, c1 = {}, c2 = {}, c3 = {};
  for (int k = 0; k < 64; k += 4) {
    const int v0i = vbase + k + 2 * hf;
    const int v1i = v0i + 1;
    const float d0 = (float)v0i - ofs - qf;
    const float d1 = (float)v1i - ofs - qf;
    float a0 = 0.3989422804014327f * __expf(-0.5f * d0 * d0);
    float a1 = 0.3989422804014327f * __expf(-0.5f * d1 * d1);
    a0 = (v0i >= 0 && v0i < Ll) ? a0 : 0.0f;   // branchless edge handling
    a1 = (v1i >= 0 && v1i < Ll) ? a1 : 0.0f;
    const int vc0 = min(max(v0i, 0), Ll - 1);
    const int vc1 = min(max(v1i, 0), Ll - 1);
    const float* r0 = Vb + (size_t)vc0 * Ee + col;
    const float* r1 = Vb + (size_t)vc1 * Ee + col;
    v2f a  = {a0, a1};
    v2f b0 = {r0[0],  r1[0]};
    v2f b1 = {r0[16], r1[16]};
    v2f b2 = {r0[32], r1[32]};
    v2f b3 = {r0[48], r1[48]};
    c0 = wmma_f32_16x16x4(a, b0, c0);
    c1 = wmma_f32_16x16x4(a, b1, c1);
    c2 = wmma_f32_16x16x4(a, b2, c2);
    c3 = wmma_f32_16x16x4(a, b3, c3);
  }

  float* Ob = O + (size_t)b * Ll * Ee;
#pragma unroll
  for (int g = 0; g < 8; ++g) {
    const int q = q0 + g + 8 * hf;
    float* orow = Ob + (size_t)q * Ee + col;
    orow[0]  = c0[g];
    orow[16] = c1[g];
    orow[32] = c2[g];
    orow[48] = c3[g];
  }
}

extern "C" void kernel_launch(void* const* d_in, const int* in_sizes, int n_in,
                              void* d_out, int out_size, void* d_ws, size_t ws_size,
                              hipStream_t stream) {
  const float* values          = (const float*)d_in[0];
  // d_in[1] (queries) is only used for its shape in the reference — unused.
  const unsigned char* keymask = (const unsigned char*)d_in[2];
  const float* w_in            = (const float*)d_in[3];
  const float* w_out           = (const float*)d_in[4];
  float* out = (float*)d_out;

  float* Y   = (float*)d_ws;                    // [B*L, E] projected values
  float* Att = Y + (size_t)Bb * Ll * Ee;        // [B*L, E] attended

  const int M = Bb * Ll;                        // 8192
  const int gemm_blocks = ((M / 32) * (Ee / 64)) / 8;   // 8 waves / block
  const int attn_blocks = (Bb * Hh * (Ll / 16) * 2) / 8;

  gemm_xwt<true ><<<gemm_blocks, 256, 0, stream>>>(values, w_in,  Y,   M, Ee, Ee, keymask);
  gauss_attn_apply<<<attn_blocks, 256, 0, stream>>>(Y, Att);
  gemm_xwt<false><<<gemm_blocks, 256, 0, stream>>>(Att,    w_out, out, M, Ee, Ee, nullptr);
}